// Attention_4080218931957
// MI455X (gfx1250) — compile-verified
//
#include <hip/hip_runtime.h>

#define B_   64
#define S_   2048
#define RNN_ 1024
#define AH_  512
#define SCHUNKS 8

typedef __attribute__((ext_vector_type(2))) float v2f;
typedef __attribute__((ext_vector_type(4))) float v4f;
typedef __attribute__((ext_vector_type(8))) float v8f;

// ---------------------------------------------------------------------------
// Kernel 1: att_h[B, AH] = h[B, RNN] @ W_h[AH, RNN]^T + b_h  via f32 WMMA.
// One wave per 16x16 output tile. A-frag 16x4: lanes 0-15 -> K=0,1 ; lanes
// 16-31 -> K=2,3 (contiguous float2 per lane). B-frag 4x16 mirrors it with
// N striped across lanes. D-frag: VGPR r -> M = r (+8 for upper half-wave).
// ---------------------------------------------------------------------------
__global__ __launch_bounds__(32) void att_h_wmma_kernel(
    const float* __restrict__ h, const float* __restrict__ W,
    const float* __restrict__ bias, float* __restrict__ att_h) {
  const int lane = threadIdx.x & 31;
  const int tileM = blockIdx.x * 16;  // batch rows
  const int tileN = blockIdx.y * 16;  // att_hid cols
  const int half = lane >> 4;         // 0: lanes 0-15, 1: lanes 16-31
  const int l15 = lane & 15;

  const float bcol = bias[tileN + l15];
  v8f acc;
#pragma unroll
  for (int r = 0; r < 8; ++r) acc[r] = bcol;

  const float* hrow = h + (size_t)(tileM + l15) * RNN_ + half * 2;
  const float* wrow = W + (size_t)(tileN + l15) * RNN_ + half * 2;

  for (int k = 0; k < RNN_; k += 4) {
    v2f a = *(const v2f*)(hrow + k);
    v2f b = *(const v2f*)(wrow + k);
    // 8 args: (neg_a, A, neg_b, B, c_mod, C, reuse_a, reuse_b)
    acc = __builtin_amdgcn_wmma_f32_16x16x4_f32(
        false, a, false, b, (short)0, acc, false, false);
  }

#pragma unroll
  for (int r = 0; r < 8; ++r) {
    const int m = tileM + r + half * 8;
    att_h[(size_t)m * AH_ + tileN + l15] = acc[r];
  }
}

// ---------------------------------------------------------------------------
// Kernel 2: scores[b,s] = sum_h tanh(p[b,s,h] + att_h[b,h]) * w_a[h] + b_a
// One wave32 per (b,s) row; att_h row + w_a staged in LDS; float4 loads.
// ---------------------------------------------------------------------------
__global__ __launch_bounds__(256) void scores_kernel(
    const float* __restrict__ p_att, const float* __restrict__ att_h,
    const float* __restrict__ w_a, const float* __restrict__ b_a,
    float* __restrict__ scores) {
  __shared__ float sAH[AH_];
  __shared__ float sWA[AH_];
  const int b = blockIdx.x;
  const int t = threadIdx.x;
  for (int i = t; i < AH_; i += 256) {
    sAH[i] = att_h[(size_t)b * AH_ + i];
    sWA[i] = w_a[i];
  }
  __syncthreads();

  const int wave = t >> 5, lane = t & 31;
  const int s = blockIdx.y * 8 + wave;
  const float* row = p_att + ((size_t)b * S_ + s) * AH_;

  float acc = 0.f;
#pragma unroll
  for (int j = 0; j < 4; ++j) {
    const int idx = j * 128 + lane * 4;
    v4f p = *(const v4f*)(row + idx);
    acc += tanhf(p.x + sAH[idx + 0]) * sWA[idx + 0];
    acc += tanhf(p.y + sAH[idx + 1]) * sWA[idx + 1];
    acc += tanhf(p.z + sAH[idx + 2]) * sWA[idx + 2];
    acc += tanhf(p.w + sAH[idx + 3]) * sWA[idx + 3];
  }
#pragma unroll
  for (int off = 16; off > 0; off >>= 1) acc += __shfl_xor(acc, off, 32);
  if (lane == 0) scores[(size_t)b * S_ + s] = acc + b_a[0];
}

// ---------------------------------------------------------------------------
// Kernel 3: masked softmax per batch row.
// softmax -> *mask -> renormalize  ==  e^{x-max}*m / sum(e^{x-max}*m)
// ---------------------------------------------------------------------------
__global__ __launch_bounds__(256) void softmax_kernel(
    const float* __restrict__ scores, const int* __restrict__ masks,
    float* __restrict__ weight) {
  __shared__ float red[256];
  const int b = blockIdx.x, t = threadIdx.x;
  float v[8], mk[8];
  float mx = -1e30f;
#pragma unroll
  for (int i = 0; i < 8; ++i) {
    const int s = t + i * 256;
    v[i] = scores[(size_t)b * S_ + s];
    mk[i] = (float)masks[(size_t)b * S_ + s];
    mx = fmaxf(mx, v[i]);
  }
  red[t] = mx;
  __syncthreads();
  for (int off = 128; off > 0; off >>= 1) {
    if (t < off) red[t] = fmaxf(red[t], red[t + off]);
    __syncthreads();
  }
  mx = red[0];
  __syncthreads();

  float sum = 0.f;
#pragma unroll
  for (int i = 0; i < 8; ++i) {
    v[i] = __expf(v[i] - mx) * mk[i];
    sum += v[i];
  }
  red[t] = sum;
  __syncthreads();
  for (int off = 128; off > 0; off >>= 1) {
    if (t < off) red[t] += red[t + off];
    __syncthreads();
  }
  const float inv = 1.0f / red[0];
#pragma unroll
  for (int i = 0; i < 8; ++i)
    weight[(size_t)b * S_ + t + i * 256] = v[i] * inv;
}

// ---------------------------------------------------------------------------
// Kernel 4: partial[b, sc, d] = sum_{s in chunk sc} weight[b,s]*att[b,s,d]
// Fully coalesced streaming of the 512 MB tensor; weights in LDS.
// ---------------------------------------------------------------------------
__global__ __launch_bounds__(256) void wsum_kernel(
    const float* __restrict__ weight, const float* __restrict__ att_feats,
    float* __restrict__ partial) {
  __shared__ float sW[256];
  const int b = blockIdx.x;   // 0..63
  const int dc = blockIdx.y;  // 0..3
  const int sc = blockIdx.z;  // 0..7
  const int t = threadIdx.x;
  const int s0 = sc * 256;
  sW[t] = weight[(size_t)b * S_ + s0 + t];
  __syncthreads();

  const int d = dc * 256 + t;
  const float* base = att_feats + ((size_t)b * S_ + s0) * RNN_ + d;
  float acc = 0.f;
#pragma unroll 4
  for (int i = 0; i < 256; ++i) acc += sW[i] * base[(size_t)i * RNN_];
  partial[((size_t)b * SCHUNKS + sc) * RNN_ + d] = acc;
}

// ---------------------------------------------------------------------------
// Kernel 5: out[b,d] = sum_sc partial[b,sc,d]
// ---------------------------------------------------------------------------
__global__ __launch_bounds__(256) void reduce_kernel(
    const float* __restrict__ partial, float* __restrict__ out) {
  const int idx = blockIdx.x * 256 + threadIdx.x;  // over B_*RNN_
  const int b = idx / RNN_, d = idx % RNN_;
  float acc = 0.f;
#pragma unroll
  for (int c = 0; c < SCHUNKS; ++c)
    acc += partial[((size_t)b * SCHUNKS + c) * RNN_ + d];
  out[idx] = acc;
}

extern "C" void kernel_launch(void* const* d_in, const int* in_sizes, int n_in,
                              void* d_out, int out_size, void* d_ws,
                              size_t ws_size, hipStream_t stream) {
  const float* h         = (const float*)d_in[0];
  const float* att_feats = (const float*)d_in[1];
  const float* p_att     = (const float*)d_in[2];
  const int*   masks     = (const int*)d_in[3];
  const float* W_h       = (const float*)d_in[4];
  const float* b_h       = (const float*)d_in[5];
  const float* w_a       = (const float*)d_in[6];
  const float* b_a       = (const float*)d_in[7];
  float* out = (float*)d_out;

  float* att_h   = (float*)d_ws;              // 64*512   = 128 KB
  float* scores  = att_h + B_ * AH_;          // 64*2048  = 512 KB
  float* weight  = scores + (size_t)B_ * S_;  // 64*2048  = 512 KB
  float* partial = weight + (size_t)B_ * S_;  // 64*8*1024 = 2 MB

  att_h_wmma_kernel<<<dim3(B_ / 16, AH_ / 16), 32, 0, stream>>>(h, W_h, b_h,
                                                                att_h);
  scores_kernel<<<dim3(B_, S_ / 8), 256, 0, stream>>>(p_att, att_h, w_a, b_a,
                                                      scores);
  softmax_kernel<<<B_, 256, 0, stream>>>(scores, masks, weight);
  wsum_kernel<<<dim3(B_, RNN_ / 256, SCHUNKS), 256, 0, stream>>>(
      weight, att_feats, partial);
  reduce_kernel<<<(B_ * RNN_) / 256, 256, 0, stream>>>(partial, out);
}